// DiffNet_Encoder_35003983462569
// MI455X (gfx1250) — compile-verified
//
#include <hip/hip_runtime.h>

typedef __attribute__((ext_vector_type(2))) float v2f;
typedef __attribute__((ext_vector_type(8))) float v8f;

#define DDIM 64

// ---------------------------------------------------------------------------
// SpMM (COO): out[row,:] += val * x[col,:]   one wave32 per edge.
// Edge metadata forced scalar (readfirstlane -> s_load); each lane gathers a
// float2 (one global_load_b64) and issues two native f32 atomics.
// ---------------------------------------------------------------------------
__global__ void spmm_edge_atomic(const int* __restrict__ rows,
                                 const int* __restrict__ cols,
                                 const float* __restrict__ vals,
                                 const float* __restrict__ x,
                                 float* __restrict__ out,
                                 int nnz) {
  int gid  = blockIdx.x * blockDim.x + threadIdx.x;
  int edge = __builtin_amdgcn_readfirstlane(gid >> 5);  // wave-uniform -> SGPR
  int lane = threadIdx.x & 31;
  if (edge >= nnz) return;                              // whole-wave uniform exit
  int   r = rows[edge];                                 // scalar loads
  int   c = cols[edge];
  float v = vals[edge];
  const float2* src = (const float2*)(x + (size_t)c * DDIM);
  float*        dst = out + (size_t)r * DDIM;
  float2 s = src[lane];                                 // one b64 gather per lane
  unsafeAtomicAdd(dst + 2 * lane,     v * s.x);         // global_atomic_add_f32
  unsafeAtomicAdd(dst + 2 * lane + 1, v * s.y);
}

// ---------------------------------------------------------------------------
// out[U,64] = [newu | uin] (U x 128) @ W (128 x 64)  via V_WMMA_F32_16X16X4_F32
// One wave -> 16-row x 64-col output stripe (4 accumulator tiles).
// W staged in LDS as K-pairs: Wp[t][n] = (W[2t][n], W[2t+1][n]) so each B
// fragment is a single aligned ds_load_b64. Pair-row stride padded to 80
// entries so lanes 0-15 / 16-31 hit disjoint bank halves (conflict-free).
// ---------------------------------------------------------------------------
#define WP_STRIDE 80

__global__ __launch_bounds__(320) void gemm_concat_wmma(
    const float* __restrict__ newu,
    const float* __restrict__ uin,
    const float* __restrict__ W,     // [128,64] row-major
    float* __restrict__ out,
    int nrows) {
  __shared__ v2f Wp[64 * WP_STRIDE];                    // 40 KB
  for (int idx = threadIdx.x; idx < 64 * 64; idx += blockDim.x) {
    int t = idx >> 6, n = idx & 63;
    v2f p;
    p.x = W[(size_t)(2 * t) * DDIM + n];
    p.y = W[(size_t)(2 * t + 1) * DDIM + n];
    Wp[t * WP_STRIDE + n] = p;
  }
  __syncthreads();

  int lane = threadIdx.x & 31;
  int tile = blockIdx.x * (blockDim.x >> 5) + (threadIdx.x >> 5);
  if (tile * 16 >= nrows) return;                       // wave-uniform
  int row0  = tile * 16;
  int m     = lane & 15;                                // A row / B col within tile
  int khalf = (lane >> 4) << 1;                         // lanes 16-31 own K = +2,+3

  const float* rowNew = newu + (size_t)(row0 + m) * DDIM;
  const float* rowU   = uin  + (size_t)(row0 + m) * DDIM;

  v8f acc0 = {}, acc1 = {}, acc2 = {}, acc3 = {};

  // K = 0..63 : A from new_u, W pair-rows t = 0..31
#pragma unroll
  for (int k0 = 0; k0 < 64; k0 += 4) {
    int kk = k0 + khalf;                                // even
    v2f a; a.x = rowNew[kk]; a.y = rowNew[kk + 1];      // global_load_b64
    const v2f* wr = &Wp[(kk >> 1) * WP_STRIDE];
    v2f b0 = wr[m];                                     // ds_load_b64 each
    v2f b1 = wr[m + 16];
    v2f b2 = wr[m + 32];
    v2f b3 = wr[m + 48];
    acc0 = __builtin_amdgcn_wmma_f32_16x16x4_f32(false, a, false, b0, (short)0, acc0, false, false);
    acc1 = __builtin_amdgcn_wmma_f32_16x16x4_f32(false, a, false, b1, (short)0, acc1, false, false);
    acc2 = __builtin_amdgcn_wmma_f32_16x16x4_f32(false, a, false, b2, (short)0, acc2, false, false);
    acc3 = __builtin_amdgcn_wmma_f32_16x16x4_f32(false, a, false, b3, (short)0, acc3, false, false);
  }

  // K = 64..127 : A from u, W pair-rows t = 32..63
#pragma unroll
  for (int k0 = 0; k0 < 64; k0 += 4) {
    int kk = k0 + khalf;
    v2f a; a.x = rowU[kk]; a.y = rowU[kk + 1];
    const v2f* wr = &Wp[(32 + (kk >> 1)) * WP_STRIDE];
    v2f b0 = wr[m];
    v2f b1 = wr[m + 16];
    v2f b2 = wr[m + 32];
    v2f b3 = wr[m + 48];
    acc0 = __builtin_amdgcn_wmma_f32_16x16x4_f32(false, a, false, b0, (short)0, acc0, false, false);
    acc1 = __builtin_amdgcn_wmma_f32_16x16x4_f32(false, a, false, b1, (short)0, acc1, false, false);
    acc2 = __builtin_amdgcn_wmma_f32_16x16x4_f32(false, a, false, b2, (short)0, acc2, false, false);
    acc3 = __builtin_amdgcn_wmma_f32_16x16x4_f32(false, a, false, b3, (short)0, acc3, false, false);
  }

  // C/D layout: VGPR v -> M = v (lanes 0-15) or v+8 (lanes 16-31), N = lane&15
  int mo = (lane >> 4) * 8;
  float* orow = out + (size_t)row0 * DDIM;
#pragma unroll
  for (int i2 = 0; i2 < 8; ++i2) {
    orow[(size_t)(mo + i2) * DDIM + m]      = acc0[i2];
    orow[(size_t)(mo + i2) * DDIM + m + 16] = acc1[i2];
    orow[(size_t)(mo + i2) * DDIM + m + 32] = acc2[i2];
    orow[(size_t)(mo + i2) * DDIM + m + 48] = acc3[i2];
  }
}

// ---------------------------------------------------------------------------
extern "C" void kernel_launch(void* const* d_in, const int* in_sizes, int n_in,
                              void* d_out, int out_size, void* d_ws, size_t ws_size,
                              hipStream_t stream) {
  const float* user_emb    = (const float*)d_in[0];
  const float* item_emb    = (const float*)d_in[1];
  const float* weights     = (const float*)d_in[2];
  const float* social_vals = (const float*)d_in[3];
  const float* inter_vals  = (const float*)d_in[4];
  const int*   social_rows = (const int*)d_in[5];
  const int*   social_cols = (const int*)d_in[6];
  const int*   inter_rows  = (const int*)d_in[7];
  const int*   inter_cols  = (const int*)d_in[8];

  const int D   = DDIM;
  const int U   = in_sizes[0] / D;
  const int I   = in_sizes[1] / D;
  const int L   = in_sizes[2] / (2 * D * D);
  const int E_S = in_sizes[3];
  const int E_I = in_sizes[4];

  float* out_u = (float*)d_out;                       // [U,64]
  float* out_i = (float*)d_out + (size_t)U * D;       // [I,64]

  float* newu = (float*)d_ws;                         // [U,64] scatter target
  float* bufA = newu + (size_t)U * D;                 // ping
  float* bufB = bufA + (size_t)U * D;                 // pong

  const int spmmThreads = 256;
  auto spmmBlocks = [&](int nnz) {
    return (int)(((size_t)nnz * 32 + spmmThreads - 1) / spmmThreads);
  };

  const int wavesPerBlock = 10;
  const int gemmThreads   = wavesPerBlock * 32;       // 320
  const int tiles         = (U + 15) / 16;
  const int gemmBlocks    = (tiles + wavesPerBlock - 1) / wavesPerBlock;

  const float* ucur = user_emb;
  for (int k = 0; k < L; ++k) {
    hipMemsetAsync(newu, 0, (size_t)U * D * sizeof(float), stream);
    spmm_edge_atomic<<<spmmBlocks(E_S), spmmThreads, 0, stream>>>(
        social_rows, social_cols, social_vals, ucur, newu, E_S);
    float* dst = (k == L - 1) ? out_u : ((k & 1) ? bufB : bufA);
    gemm_concat_wmma<<<gemmBlocks, gemmThreads, 0, stream>>>(
        newu, ucur, weights + (size_t)k * 2 * D * D, dst, U);
    ucur = dst;
  }
  if (L == 0) {
    hipMemcpyAsync(out_u, user_emb, (size_t)U * D * sizeof(float),
                   hipMemcpyDeviceToDevice, stream);
  }

  // final_u = u + spmm(inter_rows, inter_cols, inter_vals, item_emb)
  spmm_edge_atomic<<<spmmBlocks(E_I), spmmThreads, 0, stream>>>(
      inter_rows, inter_cols, inter_vals, item_emb, out_u, E_I);

  // second output: item_emb passthrough
  hipMemcpyAsync(out_i, item_emb, (size_t)I * D * sizeof(float),
                 hipMemcpyDeviceToDevice, stream);
}